// UE_Mamba_encoder_Gated_50706383897030
// MI455X (gfx1250) — compile-verified
//
#include <hip/hip_runtime.h>

typedef float v2f __attribute__((ext_vector_type(2)));
typedef float v8f __attribute__((ext_vector_type(8)));

// ---------------------------------------------------------------------------
// Channel-first LayerNorm: per (b,l) pixel, normalize across C channels.
// ---------------------------------------------------------------------------
__global__ __launch_bounds__(256)
void ln_cf_kernel(const float* __restrict__ x, long long xBS,
                  float* __restrict__ y, long long yBS,
                  const float* __restrict__ w, const float* __restrict__ bias,
                  int C, int L)
{
    int l = blockIdx.x * 256 + threadIdx.x;
    int b = blockIdx.y;
    if (l >= L) return;
    const float* xb = x + (long long)b * xBS + l;
    float mu = 0.f;
    for (int c = 0; c < C; ++c) mu += xb[(long long)c * L];
    mu /= (float)C;
    float var = 0.f;
    for (int c = 0; c < C; ++c) { float d = xb[(long long)c * L] - mu; var += d * d; }
    var /= (float)C;
    float inv = rsqrtf(var + 1e-6f);
    float* yb = y + (long long)b * yBS + l;
    for (int c = 0; c < C; ++c)
        yb[(long long)c * L] = w[c] * (xb[(long long)c * L] - mu) * inv + bias[c];
}

// ---------------------------------------------------------------------------
// 1x1-conv GEMM via V_WMMA_F32_16X16X4_F32, 4 N-tiles (16x64 output) / wave.
//   Y[b, m, l] = sum_k W[m,k] * X[b,k,l] + bias[m]   (act: 0=none, 1=sigmoid)
// A (16x4 f32, ISA 7.12.2): lanes 0-15 -> K=0,1 ; lanes 16-31 -> K=2,3.
// B (4x16 f32): rows striped across lanes (mirror of C/D layout).
// C/D (16x16 f32): VGPR r holds row M = r + 8*(lane>=16), N = lane&15.
// Out-of-range M rows: A row pointer is CLAMPED (a bad A row only corrupts the
// matching D row, which is masked at the store) -> no divergence in K-loop.
// Software-pipelined: K-step i+1 fragments load before WMMA of step i issues.
// ---------------------------------------------------------------------------
__global__ __launch_bounds__(32)
void gemm1x1_wmma(const float* __restrict__ X, long long xBS,
                  const float* __restrict__ W, const float* __restrict__ bias,
                  float* __restrict__ Y, long long yBS,
                  int M, int K, int L, int act)
{
    const int lane = threadIdx.x;        // 0..31
    const int n0   = blockIdx.x * 64;    // pixel tile (4 x 16)
    const int m0   = blockIdx.y * 16;    // out-channel tile
    const int b    = blockIdx.z;
    const int hi   = lane >> 4;          // 0 or 1
    const int l15  = lane & 15;
    const float* __restrict__ Xb = X + (long long)b * xBS;

    const int mrow = m0 + l15;
    const int arow = (mrow < M) ? mrow : (M - 1);   // clamp, no guard in loop
    int aOff = arow * K + hi * 2;
    int bOff = (hi * 2) * L + n0 + l15;
    const int L4 = 4 * L;

    v8f acc0 = {}, acc1 = {}, acc2 = {}, acc3 = {};
    v2f a, b0, b1, b2, b3;
    // prologue fragments (K >= 4 always)
    a[0]  = W[aOff];          a[1]  = W[aOff + 1];
    b0[0] = Xb[bOff     ];    b0[1] = Xb[bOff + L     ];
    b1[0] = Xb[bOff + 16];    b1[1] = Xb[bOff + L + 16];
    b2[0] = Xb[bOff + 32];    b2[1] = Xb[bOff + L + 32];
    b3[0] = Xb[bOff + 48];    b3[1] = Xb[bOff + L + 48];

    for (int k = 4; k < K; k += 4) {
        aOff += 4; bOff += L4;
        v2f an, c0, c1, c2, c3;                      // next K-step fragments
        an[0] = W[aOff];          an[1] = W[aOff + 1];
        c0[0] = Xb[bOff     ];    c0[1] = Xb[bOff + L     ];
        c1[0] = Xb[bOff + 16];    c1[1] = Xb[bOff + L + 16];
        c2[0] = Xb[bOff + 32];    c2[1] = Xb[bOff + L + 32];
        c3[0] = Xb[bOff + 48];    c3[1] = Xb[bOff + L + 48];
        acc0 = __builtin_amdgcn_wmma_f32_16x16x4_f32(false, a, false, b0, (short)0, acc0, false, false);
        acc1 = __builtin_amdgcn_wmma_f32_16x16x4_f32(false, a, false, b1, (short)0, acc1, false, false);
        acc2 = __builtin_amdgcn_wmma_f32_16x16x4_f32(false, a, false, b2, (short)0, acc2, false, false);
        acc3 = __builtin_amdgcn_wmma_f32_16x16x4_f32(false, a, false, b3, (short)0, acc3, false, false);
        a = an; b0 = c0; b1 = c1; b2 = c2; b3 = c3;
    }
    acc0 = __builtin_amdgcn_wmma_f32_16x16x4_f32(false, a, false, b0, (short)0, acc0, false, false);
    acc1 = __builtin_amdgcn_wmma_f32_16x16x4_f32(false, a, false, b1, (short)0, acc1, false, false);
    acc2 = __builtin_amdgcn_wmma_f32_16x16x4_f32(false, a, false, b2, (short)0, acc2, false, false);
    acc3 = __builtin_amdgcn_wmma_f32_16x16x4_f32(false, a, false, b3, (short)0, acc3, false, false);

#pragma unroll
    for (int r = 0; r < 8; ++r) {
        const int mr = m0 + r + 8 * hi;
        if (mr < M) {
            const float bi = bias[mr];
            float v0 = acc0[r] + bi, v1 = acc1[r] + bi, v2 = acc2[r] + bi, v3 = acc3[r] + bi;
            if (act == 1) {
                v0 = 1.f / (1.f + __expf(-v0)); v1 = 1.f / (1.f + __expf(-v1));
                v2 = 1.f / (1.f + __expf(-v2)); v3 = 1.f / (1.f + __expf(-v3));
            }
            float* yp = Y + (long long)b * yBS + (long long)mr * L + n0 + l15;
            yp[0] = v0; yp[16] = v1; yp[32] = v2; yp[48] = v3;
        }
    }
}

// ---------------------------------------------------------------------------
// Depthwise 3x3 conv, SAME padding, + bias, + SiLU.
// ---------------------------------------------------------------------------
__global__ __launch_bounds__(256)
void dwconv_silu(const float* __restrict__ x, long long xBS,
                 float* __restrict__ y, long long yBS,
                 const float* __restrict__ w, const float* __restrict__ bias,
                 int C, int H, int Wd)
{
    const int L = H * Wd;
    int t = blockIdx.x * 256 + threadIdx.x;
    int b = blockIdx.y;
    if (t >= C * L) return;
    int c = t / L, p = t % L, h = p / Wd, wc = p % Wd;
    const float* xb = x + (long long)b * xBS + (long long)c * L;
    float acc = bias[c];
#pragma unroll
    for (int dy = -1; dy <= 1; ++dy) {
        int hh = h + dy; if (hh < 0 || hh >= H) continue;
#pragma unroll
        for (int dx = -1; dx <= 1; ++dx) {
            int ww = wc + dx; if (ww < 0 || ww >= Wd) continue;
            acc += xb[hh * Wd + ww] * w[c * 9 + (dy + 1) * 3 + (dx + 1)];
        }
    }
    y[(long long)b * yBS + (long long)c * L + p] = acc / (1.f + __expf(-acc));
}

// ---------------------------------------------------------------------------
// Selective scan: 16 lanes per (b,c) channel, lane n owns state n (N=16).
//   h_n <- h_n * exp(dt*A[c,n]) + dt*u*B[b,n,t];  y = sum_n h_n*C[b,n,t] + u*D[c]
// rev=1 iterates t = L-1..0. Software-pipelined: t+1 operands load while the
// exp/shuffle chain of step t executes; periodic global_prefetch pulls the
// four streams ahead of the serial consumer.
// ---------------------------------------------------------------------------
__global__ __launch_bounds__(64)
void sel_scan(const float* __restrict__ u, long long uBS,
              const float* __restrict__ dt, long long dtBS,
              const float* __restrict__ xdbl, long long xdblBS,
              const float* __restrict__ dtbias,
              const float* __restrict__ Alog,
              const float* __restrict__ Dp,
              float* __restrict__ y, long long yBS,
              int Cch, int L, int r, int rev)
{
    const int grp = blockIdx.x * 4 + (threadIdx.x >> 4);   // (b,c) group
    const int n   = threadIdx.x & 15;
    const int b   = grp / Cch;
    const int c   = grp % Cch;
    const float A  = -__expf(Alog[c * 16 + n]);
    const float Dc = Dp[c];
    const float db = dtbias[c];
    const float* ub  = u    + (long long)b * uBS    + (long long)c * L;
    const float* dtp = dt   + (long long)b * dtBS   + (long long)c * L;
    const float* Bp  = xdbl + (long long)b * xdblBS + (long long)(r + n) * L;
    const float* Cp  = xdbl + (long long)b * xdblBS + (long long)(r + 16 + n) * L;
    float* yb = y + (long long)b * yBS + (long long)c * L;

    float h = 0.f;
    int tt = rev ? (L - 1) : 0;
    float dvC = dtp[tt], uvC = ub[tt], BvC = Bp[tt], CvC = Cp[tt];
    for (int t = 0; t < L; ++t) {
        // next-step index, clamped so the pipelined loads are always in-range
        int ttn = rev ? (L - 2 - t) : (t + 1);
        ttn = ttn < 0 ? 0 : (ttn >= L ? L - 1 : ttn);
        const float dvN = dtp[ttn], uvN = ub[ttn], BvN = Bp[ttn], CvN = Cp[ttn];
        if ((t & 31) == 0) {                         // stream prefetch
            int tp = rev ? (tt - 96) : (tt + 96);
            if (tp >= 0 && tp < L) {
                __builtin_prefetch(&Bp[tp], 0, 1);
                __builtin_prefetch(&Cp[tp], 0, 1);
                __builtin_prefetch(&ub[tp], 0, 1);
                __builtin_prefetch(&dtp[tp], 0, 1);
            }
        }
        float dv = dvC + db;
        dv = (dv > 20.f) ? dv : log1pf(__expf(dv));  // softplus
        h = h * __expf(dv * A) + dv * uvC * BvC;
        float p = h * CvC;
        p += __shfl_xor(p, 8, 16);
        p += __shfl_xor(p, 4, 16);
        p += __shfl_xor(p, 2, 16);
        p += __shfl_xor(p, 1, 16);
        if (n == 0) yb[tt] = p + uvC * Dc;
        tt = ttn; dvC = dvN; uvC = uvN; BvC = BvN; CvC = CvN;
    }
}

// ---------------------------------------------------------------------------
// Per-pixel channel mean -> sigmoid -> entropy.
// ---------------------------------------------------------------------------
__global__ __launch_bounds__(256)
void mean_entropy(const float* __restrict__ xm, long long xBS,
                  float* __restrict__ unc, int C, int L)
{
    int l = blockIdx.x * 256 + threadIdx.x;
    int b = blockIdx.y;
    if (l >= L) return;
    const float* p = xm + (long long)b * xBS + l;
    float s = 0.f;
    for (int c = 0; c < C; ++c) s += p[(long long)c * L];
    s /= (float)C;
    float sig = 1.f / (1.f + __expf(-s));
    unc[b * L + l] = -(sig * __logf(sig + 1e-6f));
}

// ---------------------------------------------------------------------------
// In-LDS bitonic argsort (descending key, stable via index tie-break).
// ---------------------------------------------------------------------------
__global__ __launch_bounds__(1024)
void argsort_desc(const float* __restrict__ unc, int* __restrict__ idx, int L)
{
    __shared__ float key[4096];
    __shared__ int   ind[4096];
    const int b = blockIdx.x, tid = threadIdx.x;
    for (int i = tid; i < L; i += 1024) { key[i] = unc[b * L + i]; ind[i] = i; }
    __syncthreads();
    for (int k = 2; k <= L; k <<= 1) {
        for (int j = k >> 1; j > 0; j >>= 1) {
            for (int i = tid; i < L; i += 1024) {
                const int ixj = i ^ j;
                if (ixj > i) {
                    const bool up = ((i & k) == 0);
                    float k1 = key[i], k2 = key[ixj];
                    int   i1 = ind[i], i2 = ind[ixj];
                    bool before = (k1 > k2) || (k1 == k2 && i1 < i2);
                    bool doswap = up ? (!before) : before;
                    if (doswap) { key[i] = k2; key[ixj] = k1; ind[i] = i2; ind[ixj] = i1; }
                }
            }
            __syncthreads();
        }
    }
    for (int i = tid; i < L; i += 1024) idx[b * L + i] = ind[i];
}

__global__ __launch_bounds__(256)
void inv_perm(const int* __restrict__ idx, int* __restrict__ inv, int L)
{
    int t = blockIdx.x * 256 + threadIdx.x;
    int b = blockIdx.y;
    if (t < L) inv[b * L + idx[b * L + t]] = t;
}

// dst[b,c,l] = src[b,c, indices[b,l]]
__global__ __launch_bounds__(256)
void gather_l(const float* __restrict__ src, long long sBS,
              float* __restrict__ dst, long long dBS,
              const int* __restrict__ indices, int C, int L)
{
    int t = blockIdx.x * 256 + threadIdx.x;
    int b = blockIdx.y;
    if (t >= C * L) return;
    int c = t / L, l = t % L;
    dst[(long long)b * dBS + (long long)c * L + l] =
        src[(long long)b * sBS + (long long)c * L + indices[b * L + l]];
}

__global__ __launch_bounds__(256)
void bn_stats(const float* __restrict__ x, long long xBS,
              float* __restrict__ mean, float* __restrict__ var,
              int B, int L)
{
    const int c = blockIdx.x;
    __shared__ float s1[256], s2[256];
    float sum = 0.f, sq = 0.f;
    for (int b = 0; b < B; ++b)
        for (int l = threadIdx.x; l < L; l += 256) {
            float v = x[(long long)b * xBS + (long long)c * L + l];
            sum += v; sq += v * v;
        }
    s1[threadIdx.x] = sum; s2[threadIdx.x] = sq;
    __syncthreads();
    for (int s = 128; s > 0; s >>= 1) {
        if (threadIdx.x < s) { s1[threadIdx.x] += s1[threadIdx.x + s]; s2[threadIdx.x] += s2[threadIdx.x + s]; }
        __syncthreads();
    }
    if (threadIdx.x == 0) {
        float m = s1[0] / (float)(B * L);
        mean[c] = m;
        var[c]  = s2[0] / (float)(B * L) - m * m;
    }
}

__global__ __launch_bounds__(256)
void bn_relu(float* __restrict__ x, long long xBS,
             const float* __restrict__ mean, const float* __restrict__ var,
             const float* __restrict__ g, const float* __restrict__ bias,
             int C, int L)
{
    int t = blockIdx.x * 256 + threadIdx.x;
    int b = blockIdx.y;
    if (t >= C * L) return;
    int c = t / L;
    long long o = (long long)b * xBS + t;
    float v = g[c] * (x[o] - mean[c]) * rsqrtf(var[c] + 1e-5f) + bias[c];
    x[o] = v > 0.f ? v : 0.f;
}

// fused = g*yu + (1-g)*concat ; concat = gcat[ch 0..C), yu = gcat[ch C..2C)
__global__ __launch_bounds__(256)
void fuse_gate(const float* __restrict__ g, long long gBS,
               const float* __restrict__ gcat, long long gcBS,
               float* __restrict__ outcat, long long oBS, int C, int L)
{
    int t = blockIdx.x * 256 + threadIdx.x;
    int b = blockIdx.y;
    if (t >= C * L) return;
    float gv = g[(long long)b * gBS + t];
    float cc = gcat[(long long)b * gcBS + t];
    float yu = gcat[(long long)b * gcBS + (long long)C * L + t];
    outcat[(long long)b * oBS + t] = gv * yu + (1.f - gv) * cc;
}

// ---------------------------------------------------------------------------
extern "C" void kernel_launch(void* const* d_in, const int* in_sizes, int n_in,
                              void* d_out, int out_size, void* d_ws, size_t ws_size,
                              hipStream_t stream)
{
    (void)in_sizes; (void)n_in; (void)out_size; (void)ws_size;
    constexpr int Bb = 2, C = 128, Hh = 64, Ww = 64, Lp = Hh * Ww, halfC = 64;

    const float* x         = (const float*)d_in[0];
    const float* ln_in_w   = (const float*)d_in[1];
    const float* ln_in_b   = (const float*)d_in[2];
    const float* inproj_w  = (const float*)d_in[3];
    const float* inproj_b  = (const float*)d_in[4];
    const float* ln_m_w    = (const float*)d_in[5];
    const float* ln_m_b    = (const float*)d_in[6];
    const float* convm_w   = (const float*)d_in[7];
    const float* convm_b   = (const float*)d_in[8];
    const float* convr_w   = (const float*)d_in[9];
    const float* convr_b   = (const float*)d_in[10];
    const float* ln_cat_w  = (const float*)d_in[11];
    const float* ln_cat_b  = (const float*)d_in[12];
    const float* xpf_w     = (const float*)d_in[13];
    const float* xpf_b     = (const float*)d_in[14];
    const float* dtf_w     = (const float*)d_in[15];
    const float* dtf_b     = (const float*)d_in[16];
    const float* Alog_f    = (const float*)d_in[17];
    const float* D_f       = (const float*)d_in[18];
    const float* xpb_w     = (const float*)d_in[19];
    const float* xpb_b     = (const float*)d_in[20];
    const float* dtb_w     = (const float*)d_in[21];
    const float* dtb_b     = (const float*)d_in[22];
    const float* Alog_b    = (const float*)d_in[23];
    const float* D_b       = (const float*)d_in[24];
    const float* xpu_w     = (const float*)d_in[25];
    const float* xpu_b     = (const float*)d_in[26];
    const float* dtu_w     = (const float*)d_in[27];
    const float* dtu_b     = (const float*)d_in[28];
    const float* Alog_u    = (const float*)d_in[29];
    const float* D_u       = (const float*)d_in[30];
    const float* gate1_w   = (const float*)d_in[31];
    const float* gate1_b   = (const float*)d_in[32];
    const float* bn_g      = (const float*)d_in[33];
    const float* bn_b      = (const float*)d_in[34];
    const float* gate2_w   = (const float*)d_in[35];
    const float* gate2_b   = (const float*)d_in[36];
    const float* outproj_w = (const float*)d_in[37];
    const float* outproj_b = (const float*)d_in[38];
    const float* outln_w   = (const float*)d_in[39];
    const float* outln_b   = (const float*)d_in[40];

    // -------- workspace bump allocator --------
    char* wp = (char*)d_ws;
    auto alloc = [&](size_t bytes) -> void* {
        void* p = (void*)wp;
        wp += (bytes + 255) & ~(size_t)255;
        return p;
    };
    const size_t CL = (size_t)C * Lp;
    float* x_ln   = (float*)alloc(Bb * CL * 4);           // (B,128,L)
    float* xp     = (float*)alloc(Bb * 2 * CL * 4);       // (B,256,L)
    float* xm_ln  = (float*)alloc(Bb * CL * 4);           // (B,128,L)
    float* xmamba = (float*)alloc(Bb * CL * 4);           // (B,128,L)
    float* outcat = (float*)alloc(Bb * 2 * CL * 4);       // (B,256,L) fused | x_res
    float* xdbl_f = (float*)alloc((size_t)Bb * 48 * Lp * 4);
    float* xdbl_b = (float*)alloc((size_t)Bb * 48 * Lp * 4);
    float* dt_f   = (float*)alloc((size_t)Bb * 64 * Lp * 4);
    float* dt_b   = (float*)alloc((size_t)Bb * 64 * Lp * 4);
    float* yfb    = (float*)alloc(Bb * CL * 4);           // yf | yb
    float* gcat   = (float*)alloc(Bb * 2 * CL * 4);       // concat_ln | yu
    float* unc    = (float*)alloc((size_t)Bb * Lp * 4);
    int*   idxb   = (int*)  alloc((size_t)Bb * Lp * 4);
    int*   invb   = (int*)  alloc((size_t)Bb * Lp * 4);
    float* xs     = (float*)alloc(Bb * CL * 4);
    float* xdbl_u = (float*)alloc((size_t)Bb * 48 * Lp * 4);
    float* dt_u   = (float*)alloc(Bb * CL * 4);
    float* yu_s   = (float*)alloc(Bb * CL * 4);
    float* g1     = (float*)alloc(Bb * CL * 4);
    float* gbuf   = (float*)alloc(Bb * CL * 4);
    float* outp   = (float*)alloc(Bb * CL * 4);
    float* bn_m   = (float*)alloc(C * 4);
    float* bn_v   = (float*)alloc(C * 4);

    const dim3 pixGrid(Lp / 256, Bb);
    const dim3 elGrid((int)((CL + 255) / 256), Bb);
    const int  NT = Lp / 64;   // GEMM pixel tiles (4x16 per wave)

    // 1) input LayerNorm
    ln_cf_kernel<<<pixGrid, 256, 0, stream>>>(x, (long long)CL, x_ln, (long long)CL,
                                              ln_in_w, ln_in_b, C, Lp);
    // 2) inproj: 256x128 GEMM  (WMMA f32)
    gemm1x1_wmma<<<dim3(NT, 16, Bb), 32, 0, stream>>>(
        x_ln, (long long)CL, inproj_w, inproj_b, xp, (long long)(2 * CL), 256, 128, Lp, 0);
    // 3) ln_mamba on first 128 channels of xp
    ln_cf_kernel<<<pixGrid, 256, 0, stream>>>(xp, (long long)(2 * CL), xm_ln, (long long)CL,
                                              ln_m_w, ln_m_b, C, Lp);
    // 4) dwconv + silu -> x_mamba
    dwconv_silu<<<elGrid, 256, 0, stream>>>(xm_ln, (long long)CL, xmamba, (long long)CL,
                                            convm_w, convm_b, C, Hh, Ww);
    // 5) dwconv + silu on xp[:,128:] -> x_res (outcat channels 128..255)
    dwconv_silu<<<elGrid, 256, 0, stream>>>(xp + CL, (long long)(2 * CL), outcat + CL,
                                            (long long)(2 * CL), convr_w, convr_b, C, Hh, Ww);
    // 6) x_dbl projections (M=36 padded to 48)
    gemm1x1_wmma<<<dim3(NT, 3, Bb), 32, 0, stream>>>(
        xmamba, (long long)CL, xpf_w, xpf_b, xdbl_f, (long long)48 * Lp, 36, 64, Lp, 0);
    gemm1x1_wmma<<<dim3(NT, 3, Bb), 32, 0, stream>>>(
        xmamba + (size_t)halfC * Lp, (long long)CL, xpb_w, xpb_b,
        xdbl_b, (long long)48 * Lp, 36, 64, Lp, 0);
    // 7) dt projections (K=4)
    gemm1x1_wmma<<<dim3(NT, 4, Bb), 32, 0, stream>>>(
        xdbl_f, (long long)48 * Lp, dtf_w, dtf_b, dt_f, (long long)64 * Lp, 64, 4, Lp, 0);
    gemm1x1_wmma<<<dim3(NT, 4, Bb), 32, 0, stream>>>(
        xdbl_b, (long long)48 * Lp, dtb_w, dtb_b, dt_b, (long long)64 * Lp, 64, 4, Lp, 0);
    // 8) selective scans (forward / spatially-reversed)
    sel_scan<<<(Bb * halfC) / 4, 64, 0, stream>>>(
        xmamba, (long long)CL, dt_f, (long long)64 * Lp, xdbl_f, (long long)48 * Lp,
        dtf_b, Alog_f, D_f, yfb, (long long)CL, halfC, Lp, 4, 0);
    sel_scan<<<(Bb * halfC) / 4, 64, 0, stream>>>(
        xmamba + (size_t)halfC * Lp, (long long)CL, dt_b, (long long)64 * Lp,
        xdbl_b, (long long)48 * Lp, dtb_b, Alog_b, D_b,
        yfb + (size_t)halfC * Lp, (long long)CL, halfC, Lp, 4, 1);
    // 9) ln_cat([yf;yb]) -> gcat channels 0..127
    ln_cf_kernel<<<pixGrid, 256, 0, stream>>>(yfb, (long long)CL, gcat, (long long)(2 * CL),
                                              ln_cat_w, ln_cat_b, C, Lp);
    // 10-12) uncertainty ordering
    mean_entropy<<<pixGrid, 256, 0, stream>>>(xmamba, (long long)CL, unc, C, Lp);
    argsort_desc<<<Bb, 1024, 0, stream>>>(unc, idxb, Lp);
    inv_perm<<<pixGrid, 256, 0, stream>>>(idxb, invb, Lp);
    // 13) gather x_mamba by idx
    gather_l<<<elGrid, 256, 0, stream>>>(xmamba, (long long)CL, xs, (long long)CL, idxb, C, Lp);
    // 14-15) uncertainty-branch projections (M=40 pad 48, dt K=8)
    gemm1x1_wmma<<<dim3(NT, 3, Bb), 32, 0, stream>>>(
        xs, (long long)CL, xpu_w, xpu_b, xdbl_u, (long long)48 * Lp, 40, 128, Lp, 0);
    gemm1x1_wmma<<<dim3(NT, 8, Bb), 32, 0, stream>>>(
        xdbl_u, (long long)48 * Lp, dtu_w, dtu_b, dt_u, (long long)CL, 128, 8, Lp, 0);
    // 16) uncertainty scan (r=8)
    sel_scan<<<(Bb * C) / 4, 64, 0, stream>>>(
        xs, (long long)CL, dt_u, (long long)CL, xdbl_u, (long long)48 * Lp,
        dtu_b, Alog_u, D_u, yu_s, (long long)CL, C, Lp, 8, 0);
    // 17) un-permute: yu = yu_s gathered by inv -> gcat channels 128..255
    gather_l<<<elGrid, 256, 0, stream>>>(yu_s, (long long)CL, gcat + CL, (long long)(2 * CL),
                                         invb, C, Lp);
    // 18) gate1: 128x256 GEMM
    gemm1x1_wmma<<<dim3(NT, 8, Bb), 32, 0, stream>>>(
        gcat, (long long)(2 * CL), gate1_w, gate1_b, g1, (long long)CL, 128, 256, Lp, 0);
    // 19) BatchNorm (train stats) + ReLU, in place
    bn_stats<<<C, 256, 0, stream>>>(g1, (long long)CL, bn_m, bn_v, Bb, Lp);
    bn_relu<<<elGrid, 256, 0, stream>>>(g1, (long long)CL, bn_m, bn_v, bn_g, bn_b, C, Lp);
    // 20) gate2: 128x128 GEMM + sigmoid
    gemm1x1_wmma<<<dim3(NT, 8, Bb), 32, 0, stream>>>(
        g1, (long long)CL, gate2_w, gate2_b, gbuf, (long long)CL, 128, 128, Lp, 1);
    // 21) fuse -> outcat channels 0..127 (x_res already in 128..255)
    fuse_gate<<<elGrid, 256, 0, stream>>>(gbuf, (long long)CL, gcat, (long long)(2 * CL),
                                          outcat, (long long)(2 * CL), C, Lp);
    // 22) outproj: 128x256 GEMM
    gemm1x1_wmma<<<dim3(NT, 8, Bb), 32, 0, stream>>>(
        outcat, (long long)(2 * CL), outproj_w, outproj_b, outp, (long long)CL, 128, 256, Lp, 0);
    // 23) output LayerNorm -> d_out
    ln_cf_kernel<<<pixGrid, 256, 0, stream>>>(outp, (long long)CL, (float*)d_out, (long long)CL,
                                              outln_w, outln_b, C, Lp);
}